// HungarianMatcher_42279658061995
// MI455X (gfx1250) — compile-verified
//
#include <hip/hip_runtime.h>
#include <cstdint>

#define MAXV 8
#define MM 256          // square (padded) assignment size, == M in the reference
#define KS 8            // MM / 32 lanes

// ---------------- Phase 1: rotated-GIoU cost matrix ----------------

__device__ __forceinline__ void rot_corners(float cx, float cy, float w, float h, float t,
                                            float* x, float* y) {
    float c = cosf(t), s = sinf(t);
    const float dxs[4] = {-0.5f, 0.5f, 0.5f, -0.5f};
    const float dys[4] = {-0.5f, -0.5f, 0.5f, 0.5f};
#pragma unroll
    for (int k = 0; k < 4; ++k) {
        float dx = dxs[k] * w, dy = dys[k] * h;
        x[k] = cx + dx * c - dy * s;
        y[k] = cy + dx * s + dy * c;
    }
}

// GIoU cost; tx/ty = target corners (clip edges), px0/py0 = pred corners (clipped poly)
__device__ float giou_cost(const float* tx, const float* ty,
                           const float* px0, const float* py0,
                           float area_t, float area_p) {
    float px[MAXV], py[MAXV];
    int cnt = 4;
    for (int k = 0; k < 4; ++k) { px[k] = px0[k]; py[k] = py0[k]; }

    for (int e = 0; e < 4; ++e) {  // Sutherland-Hodgman: clip by target edge e
        float ax = tx[e], ay = ty[e];
        float bx = tx[(e + 1) & 3], by = ty[(e + 1) & 3];
        float dx = bx - ax, dy = by - ay;
        float qx[MAXV], qy[MAXV];
        int qc = 0;
        for (int k = 0; k < cnt && k < MAXV; ++k) {
            int kn = (k + 1 < cnt) ? k + 1 : 0;
            float sc = dx * (py[k]  - ay) - dy * (px[k]  - ax);
            float sn = dx * (py[kn] - ay) - dy * (px[kn] - ax);
            bool ic = (sc >= 0.f), in2 = (sn >= 0.f);
            if (ic) { if (qc < MAXV) { qx[qc] = px[k]; qy[qc] = py[k]; } ++qc; }
            if (ic != in2) {
                float den = sc - sn;
                den = (fabsf(den) > 1e-9f) ? den : 1e-9f;  // match reference clamp
                float tt = sc / den;
                if (qc < MAXV) {
                    qx[qc] = px[k] + tt * (px[kn] - px[k]);
                    qy[qc] = py[k] + tt * (py[kn] - py[k]);
                }
                ++qc;
            }
        }
        cnt = (qc < MAXV) ? qc : MAXV;   // convex quad intersection: never > 8
        for (int k = 0; k < MAXV; ++k) {
            px[k] = (k < cnt) ? qx[k] : 0.f;
            py[k] = (k < cnt) ? qy[k] : 0.f;
        }
    }
    float acc = 0.f;                      // shoelace
    for (int k = 0; k < MAXV; ++k) {
        if (k < cnt) {
            int kn = (k + 1 < cnt) ? k + 1 : 0;
            acc += px[k] * py[kn] - px[kn] * py[k];
        }
    }
    float inter = 0.5f * fabsf(acc);
    float uni = area_t + area_p - inter;
    float xmn = tx[0], xmx = tx[0], ymn = ty[0], ymx = ty[0];
#pragma unroll
    for (int k = 1; k < 4; ++k) {
        xmn = fminf(xmn, tx[k]); xmx = fmaxf(xmx, tx[k]);
        ymn = fminf(ymn, ty[k]); ymx = fmaxf(ymx, ty[k]);
    }
#pragma unroll
    for (int k = 0; k < 4; ++k) {
        xmn = fminf(xmn, px0[k]); xmx = fmaxf(xmx, px0[k]);
        ymn = fminf(ymn, py0[k]); ymx = fmaxf(ymx, py0[k]);
    }
    float enc = (xmx - xmn) * (ymx - ymn);
    float giou = inter / uni - (enc - uni) / enc;
    return 1.f - giou;
}

// grid = (M, B), block = 256 threads: blockIdx.x = pred row m (uniform -> SGPR loads),
// blockIdx.y = batch b, threadIdx.x = target col n. No integer div/mod anywhere.
__global__ __launch_bounds__(MM) void cost_kernel(
    const float* __restrict__ pbox, const float* __restrict__ pscore,
    const float* __restrict__ tbox, const float* __restrict__ tscore,
    float* __restrict__ cost, int M, int N) {
    const int m = blockIdx.x;
    const int b = blockIdx.y;
    const int n = threadIdx.x;

    float c = 0.f;                 // zero padding for n >= N
    if (n < N) {
        // pred box: block-uniform -> scalar loads
        const float* bp = pbox + (size_t)(b * M + m) * 5;
        float pcx = bp[0], pcy = bp[1], pw = bp[2], ph = bp[3], pt = bp[4];
        float ps = pscore[b * M + m];
        // target box: per-lane
        const float* bt = tbox + (size_t)(b * N + n) * 5;
        float tcx = bt[0], tcy = bt[1], tw = bt[2], th = bt[3], tt = bt[4];
        float ts = tscore[b * N + n];

        float txc[4], tyc[4], pxc[4], pyc[4];
        rot_corners(tcx, tcy, tw, th, tt, txc, tyc);
        rot_corners(pcx, pcy, pw, ph, pt, pxc, pyc);

        float sc = fmaxf(0.f, ts - ps);
        c = sc + giou_cost(txc, tyc, pxc, pyc, tw * th, pw * ph);  // COST_CLASS=COST_GIOU=1
    }
    cost[((size_t)b * M + m) * M + n] = c;
}

// ---------------- Phase 2+3: Hungarian (JV) per batch, one wave32 ----------------
// Whole 256x256 f32 cost matrix resident in CDNA5's 320KB LDS; staged via
// async global->LDS DMA (ASYNCcnt). Columns strided across 32 lanes, 8 per lane.

__global__ __launch_bounds__(256) void hungarian_kernel(
    const float* __restrict__ cost, const float* __restrict__ pscore,
    const float* __restrict__ tscore, float* __restrict__ out,
    int B, int N) {
    extern __shared__ float s_cost[];          // MM*MM floats = 256 KB
    __shared__ double s_u[MM + 1];
    __shared__ int s_p[MM + 1];
    __shared__ int s_way[MM + 1];

    const int b = blockIdx.x;
    const int tid = threadIdx.x;

    // async DMA: 256KB global -> LDS (CDNA5 GLOBAL_LOAD_ASYNC_TO_LDS_B128)
    {
        const float* src = cost + (size_t)b * MM * MM;
        for (int it = tid; it < (MM * MM) / 4; it += (int)blockDim.x) {
            uint32_t lds_off = (uint32_t)(uintptr_t)(const void*)&s_cost[it * 4];
            uint64_t gaddr = (uint64_t)(uintptr_t)(src + (size_t)it * 4);
            asm volatile("global_load_async_to_lds_b128 %0, %1, off"
                         :: "v"(lds_off), "v"(gaddr) : "memory");
        }
        asm volatile("s_wait_asynccnt 0" ::: "memory");
    }
    for (int idx = tid; idx <= MM; idx += (int)blockDim.x) {
        s_u[idx] = 0.0; s_p[idx] = 0; s_way[idx] = 0;
    }
    __syncthreads();
    if (tid >= 32) return;                     // wave 0 runs the serial solve
    const int lane = tid;

    volatile double* vu = s_u;
    volatile int* vp = s_p;
    volatile int* vw = s_way;

    double vloc[KS], minv[KS];
#pragma unroll
    for (int k = 0; k < KS; ++k) vloc[k] = 0.0;
    const double INF = 1e30;

    for (int i = 1; i <= MM; ++i) {
#pragma unroll
        for (int k = 0; k < KS; ++k) minv[k] = INF;
        unsigned usedm = 0;                    // per-lane used bits, cols j=1+lane+32k
        if (lane == 0) vp[0] = i;
        __builtin_amdgcn_wave_barrier();
        int j0 = 0;
        for (;;) {
            if (j0 > 0) {                      // mark used[j0]
                int jj = j0 - 1;
                if ((jj & 31) == lane) usedm |= 1u << (jj >> 5);
            }
            int i0 = vp[j0];                   // uniform LDS broadcast
            double ui0 = vu[i0];
            const float* crow = &s_cost[(i0 - 1) * MM];
            double best = INF; int bestj = 0;
#pragma unroll
            for (int k = 0; k < KS; ++k) {
                if (!((usedm >> k) & 1u)) {
                    int j = 1 + lane + 32 * k;
                    double cur = (double)crow[j - 1] - ui0 - vloc[k];
                    if (cur < minv[k]) { minv[k] = cur; vw[j] = j0; }
                    if (minv[k] < best) { best = minv[k]; bestj = j; }
                }
            }
            // wave32 argmin butterfly (tie-break: smaller column index)
#pragma unroll
            for (int off = 16; off > 0; off >>= 1) {
                double ov = __shfl_xor(best, off, 32);
                int oj = __shfl_xor(bestj, off, 32);
                if (ov < best || (ov == best && oj < bestj)) { best = ov; bestj = oj; }
            }
            double delta = best;
            int j1 = bestj;
            // dual updates: used -> u[p[j]] += delta, v[j] -= delta; free -> minv -= delta
#pragma unroll
            for (int k = 0; k < KS; ++k) {
                int j = 1 + lane + 32 * k;
                if ((usedm >> k) & 1u) {
                    int pj = vp[j];            // distinct rows across used cols
                    vu[pj] = vu[pj] + delta;
                    vloc[k] -= delta;
                } else {
                    minv[k] -= delta;
                }
            }
            if (lane == 0) vu[i] = vu[i] + delta;   // column 0 always used, p[0]=i
            __builtin_amdgcn_wave_barrier();
            j0 = j1;
            if (vp[j0] == 0) break;
        }
        // augment along alternating path
        while (j0 != 0) {
            int j1 = vw[j0];
            int pj1 = vp[j1];
            __builtin_amdgcn_wave_barrier();
            if (lane == 0) vp[j0] = pj1;
            __builtin_amdgcn_wave_barrier();
            j0 = j1;
        }
    }

    // matched-cost reductions: match, score part, box part (box = match - score)
    float msum = 0.f, ssum = 0.f;
#pragma unroll
    for (int k = 0; k < KS; ++k) {
        int j = 1 + lane + 32 * k;
        int r = vp[j] - 1;                     // row matched to column j-1
        int c = j - 1;
        msum += s_cost[r * MM + c];
        if (c < N) ssum += fmaxf(0.f, tscore[b * N + c] - pscore[b * MM + r]);
    }
#pragma unroll
    for (int off = 16; off > 0; off >>= 1) {
        msum += __shfl_xor(msum, off, 32);
        ssum += __shfl_xor(ssum, off, 32);
    }
    if (lane == 0) {
        float invN = 1.f / (float)N;
        out[0 * B + b] = msum * invN;
        out[1 * B + b] = ssum * invN;
        out[2 * B + b] = (msum - ssum) * invN;
    }
}

// ---------------- launcher ----------------

extern "C" void kernel_launch(void* const* d_in, const int* in_sizes, int n_in,
                              void* d_out, int out_size, void* d_ws, size_t ws_size,
                              hipStream_t stream) {
    const float* pbox   = (const float*)d_in[0];   // [B,M,5]
    const float* pscore = (const float*)d_in[1];   // [B,M]
    const float* tbox   = (const float*)d_in[2];   // [B,N,5]
    const float* tscore = (const float*)d_in[3];   // [B,N]
    float* out = (float*)d_out;                    // [3,B]
    float* cost_ws = (float*)d_ws;                 // B*M*M floats (1 MB)

    int B = out_size / 3;              // 4
    int M = in_sizes[1] / B;           // 256 (== MM)
    int N = in_sizes[3] / B;           // 192

    cost_kernel<<<dim3(M, B), MM, 0, stream>>>(
        pbox, pscore, tbox, tscore, cost_ws, M, N);

    size_t shmem = (size_t)M * M * sizeof(float);  // 256 KB dynamic LDS (CDNA5: 320KB/WGP)
    hungarian_kernel<<<B, 256, shmem, stream>>>(cost_ws, pscore, tscore, out, B, N);
}